// DeformableAttention_40020505264290
// MI455X (gfx1250) — compile-verified
//
#include <hip/hip_runtime.h>
#include <math.h>

// ---------------------------------------------------------------------------
// Deformable attention, MI455X (gfx1250), wave32.
// GEMMs on the matrix pipe via V_WMMA_F32_16X16X4_F32 (fp32-exact vs the
// reference; problem is L2/latency bound so low precision buys nothing).
// Weight panels staged global->LDS with GLOBAL_LOAD_ASYNC_TO_LDS_B128
// (ASYNCcnt path), shared by all 8 waves of a block.
// ---------------------------------------------------------------------------

typedef __attribute__((ext_vector_type(2))) float v2f;
typedef __attribute__((ext_vector_type(8))) float v8f;

#define EMBED_DIM 256
#define NUM_HEADS 8
#define NUM_POINTS 4
#define BATCH 2
#define NQ 2048
#define VH 100
#define VW 100

// ---------------------------------------------------------------------------
// fp32 WMMA GEMM:  C[M,N] = A[M,K] @ B[K,N] + bias[N]     (N compile-time)
// Block = 256 threads = 8 waves; block computes a 128x16 C tile (8 m-tiles x
// one shared n-tile). B panel (KC x 16) staged in LDS via async b128 copies.
// Inner loop: v_wmma_f32_16x16x4_f32, K step 4.
//
// A frag (32-bit 16x4, ISA 7.12.2): lanes 0-15: M=lane, elems K,K+1;
// lanes 16-31: M=lane-16, elems K+2,K+3. B mirrors with N across lanes.
// C/D: VGPR i -> rows i / i+8, col = lane&15.
// LDS B layout: row-major [KC][16] floats -> read pattern is bank-conflict
// free (cols hit 16 distinct banks; half-wave K-offset shifts by 32 banks).
// ---------------------------------------------------------------------------
template <int N>
__global__ __launch_bounds__(256) void wmma_gemm_f32_lds(
    const float* __restrict__ A, const float* __restrict__ Bm,
    const float* __restrict__ bias, float* __restrict__ C,
    int M, int K)
{
    constexpr int KC = 128;                 // K chunk staged per barrier
    __shared__ float Bs[KC * 16];           // 8 KB

    const int tid  = threadIdx.x;
    const int lane = tid & 31;
    const int wv   = tid >> 5;
    const int tn   = blockIdx.y;
    const int Mtiles = M >> 4;

    int tm = blockIdx.x * 8 + wv;
    if (tm >= Mtiles) tm = Mtiles - 1;      // tail clamp: keep waves converged

    const int half = lane >> 4;             // 0: K,K+1 | 1: K+2,K+3
    const int l15  = lane & 15;

    const float* __restrict__ Arow   = A  + (size_t)(tm * 16 + l15) * K;
    const float* __restrict__ Bpanel = Bm + tn * 16;

    // Cooperative staging coords: 2 threads per B row, 8 floats (32B) each.
    const int br = tid >> 1;                // 0..127
    const int bc = (tid & 1) * 8;           // 0 or 8
    const uint32_t lds_dst =
        (uint32_t)(uintptr_t)(&Bs[br * 16 + bc]);   // low 32 bits = LDS offset

    v8f acc = {};
    for (int kb = 0; kb < K; kb += KC) {
        const float* gsrc = Bpanel + (size_t)(kb + br) * N + bc;
        // 32B per thread global->LDS, no VGPR round trip (ASYNCcnt).
        asm volatile("global_load_async_to_lds_b128 %0, %1, off"
                     :: "v"(lds_dst), "v"(gsrc) : "memory");
        asm volatile("global_load_async_to_lds_b128 %0, %1, off offset:16"
                     :: "v"(lds_dst), "v"(gsrc) : "memory");

        // Prefetch next A chunk while the async copy is in flight.
        __builtin_prefetch(Arow + kb + KC, 0, 3);

        asm volatile("s_wait_asynccnt 0x0" ::: "memory");
        __syncthreads();

#pragma unroll
        for (int kk = 0; kk < KC; kk += 4) {
            const int k0 = kk + half * 2;
            v2f a = *(const v2f*)(Arow + kb + k0);      // 8B contiguous/lane
            v2f b;
            b.x = Bs[k0 * 16 + l15];                    // ds_load_b32 x2
            b.y = Bs[(k0 + 1) * 16 + l15];
            acc = __builtin_amdgcn_wmma_f32_16x16x4_f32(
                false, a, false, b, (short)0, acc, false, false);
        }
        __syncthreads();
    }

    const float bv = bias[tn * 16 + l15];
    float* __restrict__ Crow =
        C + (size_t)(tm * 16 + half * 8) * N + tn * 16 + l15;
#pragma unroll
    for (int i = 0; i < 8; ++i)
        Crow[(size_t)i * N] = acc[i] + bv;              // immediate offsets
}

// ---------------------------------------------------------------------------
// Softmax over the 4 sampling points of each (b,q,head). In-place.
// ---------------------------------------------------------------------------
__global__ __launch_bounds__(256) void softmax4_kernel(float* __restrict__ attn,
                                                       int groups)
{
    const int g = blockIdx.x * blockDim.x + threadIdx.x;
    if (g >= groups) return;
    float* a = attn + (size_t)g * 4;
    float a0 = a[0], a1 = a[1], a2 = a[2], a3 = a[3];
    float m = fmaxf(fmaxf(a0, a1), fmaxf(a2, a3));
    float e0 = __expf(a0 - m), e1 = __expf(a1 - m);
    float e2 = __expf(a2 - m), e3 = __expf(a3 - m);
    float inv = 1.0f / (e0 + e1 + e2 + e3);
    a[0] = e0 * inv; a[1] = e1 * inv; a[2] = e2 * inv; a[3] = e3 * inv;
}

// ---------------------------------------------------------------------------
// Bilinear sampling + softmax-weighted aggregation.
// One wave per (b,q,head); each lane owns 8 channels. grid_sample
// (align_corners=False, zeros) simplifies to x = ref_x*W + off_x - 0.5.
// Corner reads are 1KB contiguous rows from the L2-resident val_proj
// (2x global_load_b128 per lane).
// ---------------------------------------------------------------------------
__global__ __launch_bounds__(256) void sample_aggregate_kernel(
    const float* __restrict__ val,    // [B, H*W, D]
    const float* __restrict__ refp,   // [B*Q, 2]
    const float* __restrict__ offs,   // [B*Q, 64]
    const float* __restrict__ attn,   // [B*Q, 32] (softmaxed)
    float* __restrict__ weighted)     // [B*Q, 8*256]
{
    const int lane = threadIdx.x & 31;
    const int wave = (blockIdx.x * (blockDim.x >> 5)) + (threadIdx.x >> 5);
    const int h  = wave & (NUM_HEADS - 1);
    const int bq = wave >> 3;
    if (bq >= BATCH * NQ) return;
    const int b = bq >> 11;                       // NQ = 2048

    const float rx = refp[(size_t)bq * 2 + 0];
    const float ry = refp[(size_t)bq * 2 + 1];
    const float* __restrict__ vbase = val + (size_t)b * (VH * VW) * EMBED_DIM;
    const float* __restrict__ orow  = offs + (size_t)bq * (NUM_HEADS * NUM_POINTS * 2);
    const float* __restrict__ arow  = attn + (size_t)bq * (NUM_HEADS * NUM_POINTS);

    float acc[8];
#pragma unroll
    for (int i = 0; i < 8; ++i) acc[i] = 0.0f;

#pragma unroll
    for (int p = 0; p < NUM_POINTS; ++p) {
        const float ox = orow[(h * NUM_POINTS + p) * 2 + 0];
        const float oy = orow[(h * NUM_POINTS + p) * 2 + 1];
        const float aw = arow[h * NUM_POINTS + p];

        const float x = rx * (float)VW + ox - 0.5f;
        const float y = ry * (float)VH + oy - 0.5f;
        const float x0f = floorf(x), y0f = floorf(y);
        const int   x0 = (int)x0f,   y0 = (int)y0f;
        const float wx = x - x0f,    wy = y - y0f;

#pragma unroll
        for (int c = 0; c < 4; ++c) {
            const int dy = c >> 1, dx = c & 1;
            const int xi = x0 + dx, yi = y0 + dy;
            const bool valid = (xi >= 0) & (xi < VW) & (yi >= 0) & (yi < VH);
            const float wcorner = (dy ? wy : 1.0f - wy) * (dx ? wx : 1.0f - wx);
            const float wgt = valid ? aw * wcorner : 0.0f;
            const int xc = xi < 0 ? 0 : (xi >= VW ? VW - 1 : xi);
            const int yc = yi < 0 ? 0 : (yi >= VH ? VH - 1 : yi);
            const float4* __restrict__ row =
                (const float4*)(vbase + (size_t)(yc * VW + xc) * EMBED_DIM);
            const float4 v0 = row[lane * 2 + 0];
            const float4 v1 = row[lane * 2 + 1];
            acc[0] += wgt * v0.x; acc[1] += wgt * v0.y;
            acc[2] += wgt * v0.z; acc[3] += wgt * v0.w;
            acc[4] += wgt * v1.x; acc[5] += wgt * v1.y;
            acc[6] += wgt * v1.z; acc[7] += wgt * v1.w;
        }
    }

    float* __restrict__ out =
        weighted + ((size_t)bq * NUM_HEADS + h) * EMBED_DIM + lane * 8;
    float4* o4 = (float4*)out;
    o4[0] = make_float4(acc[0], acc[1], acc[2], acc[3]);
    o4[1] = make_float4(acc[4], acc[5], acc[6], acc[7]);
}

// ---------------------------------------------------------------------------
// Host-side launch
// ---------------------------------------------------------------------------
extern "C" void kernel_launch(void* const* d_in, const int* in_sizes, int n_in,
                              void* d_out, int out_size, void* d_ws, size_t ws_size,
                              hipStream_t stream) {
    (void)in_sizes; (void)n_in; (void)out_size; (void)ws_size;

    const float* query  = (const float*)d_in[0];   // [B, Q, D]
    const float* refp   = (const float*)d_in[1];   // [B, Q, 2]
    const float* value  = (const float*)d_in[2];   // [B, H, W, D]
    const float* W_off  = (const float*)d_in[3];   // [D, 64]
    const float* b_off  = (const float*)d_in[4];   // [64]
    const float* W_attn = (const float*)d_in[5];   // [D, 32]
    const float* b_attn = (const float*)d_in[6];   // [32]
    const float* W_v    = (const float*)d_in[7];   // [D, D]
    const float* b_v    = (const float*)d_in[8];   // [D]
    const float* W_o    = (const float*)d_in[9];   // [8*D, D]
    const float* b_o    = (const float*)d_in[10];  // [D]
    float* out = (float*)d_out;

    const int BQ   = BATCH * NQ;                   // 4096
    const int MVAL = BATCH * VH * VW;              // 20000

    // Workspace carve-up (floats): ~56 MB total.
    float* ws       = (float*)d_ws;
    float* val_proj = ws;                                                   // 20000*256
    float* offsets  = val_proj + (size_t)MVAL * EMBED_DIM;                  // 4096*64
    float* attn     = offsets  + (size_t)BQ * (NUM_HEADS * NUM_POINTS * 2); // 4096*32
    float* weighted = attn     + (size_t)BQ * (NUM_HEADS * NUM_POINTS);     // 4096*2048

    // 1) val_proj = value @ W_v + b_v    (M=20000, N=256, K=256)
    {
        dim3 grid((MVAL / 16 + 7) / 8, EMBED_DIM / 16);   // 157 x 16
        wmma_gemm_f32_lds<EMBED_DIM><<<grid, 256, 0, stream>>>(
            value, W_v, b_v, val_proj, MVAL, EMBED_DIM);
    }
    // 2) offsets = query @ W_off + b_off (M=4096, N=64, K=256)
    {
        dim3 grid((BQ / 16) / 8, (NUM_HEADS * NUM_POINTS * 2) / 16);  // 32 x 4
        wmma_gemm_f32_lds<NUM_HEADS * NUM_POINTS * 2><<<grid, 256, 0, stream>>>(
            query, W_off, b_off, offsets, BQ, EMBED_DIM);
    }
    // 3) attn logits = query @ W_attn + b_attn (M=4096, N=32, K=256)
    {
        dim3 grid((BQ / 16) / 8, (NUM_HEADS * NUM_POINTS) / 16);      // 32 x 2
        wmma_gemm_f32_lds<NUM_HEADS * NUM_POINTS><<<grid, 256, 0, stream>>>(
            query, W_attn, b_attn, attn, BQ, EMBED_DIM);
    }
    // 4) softmax over the 4 points of each (b,q,head)
    {
        int groups = BQ * NUM_HEADS;               // 32768
        softmax4_kernel<<<(groups + 255) / 256, 256, 0, stream>>>(attn, groups);
    }
    // 5) bilinear sample + weighted aggregation -> weighted [4096, 2048]
    {
        int waves = BQ * NUM_HEADS;                // 32768 waves
        sample_aggregate_kernel<<<waves / 8, 256, 0, stream>>>(
            val_proj, refp, offsets, attn, weighted);
    }
    // 6) out = weighted @ W_o + b_o      (M=4096, N=256, K=2048)
    {
        dim3 grid((BQ / 16) / 8, EMBED_DIM / 16);  // 32 x 16
        wmma_gemm_f32_lds<EMBED_DIM><<<grid, 256, 0, stream>>>(
            weighted, W_o, b_o, out, BQ, NUM_HEADS * EMBED_DIM);
    }
}